// GNNScoutPolicy_88991722373464
// MI455X (gfx1250) — compile-verified
//
#include <hip/hip_runtime.h>
#include <hip/hip_bf16.h>
#include <math.h>

// ---------------------------------------------------------------------------
// GNN scout policy on MI455X (gfx1250).
//
// Roofline: ~1.3 GFLOP total, ~50 MB of unavoidable traffic -> microseconds at
// 23.3 TB/s. Bandwidth-bound, so the design minimizes HBM movement:
//  * GATv2 layers: one block per batch; xl/xr/acc (3 x 80KB) + per-node
//    head max/sum (32KB) live entirely in the 320KB CDNA5 LDS. Segment
//    softmax via LDS atomics (ds_max via monotone-uint trick, ds_add_f32).
//  * Layer-1 input staging uses the Tensor Data Mover: one tensor_load_to_lds
//    per block pulls the 80KB [N,HD] batch tile into LDS (TENSORcnt-tracked),
//    BN0 affine applied in-place in LDS.
//  * h1 never hits HBM: BN is affine per feature, so mean(bn(h)) = bn(mean(h));
//    layer 1 emits only per-batch node sums + per-block BN partials.
//  * MLP GEMMs use V_WMMA_F32_16X16X4_F32 (full fp32, matches reference).
// ---------------------------------------------------------------------------

#define Bq   256
#define Nq   1024
#define Eq   8192
#define FINq 8
#define Hq   4
#define Dq   5
#define HDq  20
#define ESq  (Eq + Nq)      // edges + self loops
#define NTHREADS 256

typedef float    v2f __attribute__((ext_vector_type(2)));
typedef float    v8f __attribute__((ext_vector_type(8)));
typedef unsigned v4u __attribute__((ext_vector_type(4)));
typedef int      v4i __attribute__((ext_vector_type(4)));
typedef int      v8i __attribute__((ext_vector_type(8)));

// monotone float<->uint map so atomicMax(uint) implements float max
__device__ __forceinline__ unsigned fkey(float f) {
  unsigned u = __float_as_uint(f);
  return (u & 0x80000000u) ? ~u : (u | 0x80000000u);
}
__device__ __forceinline__ float finv(unsigned k) {
  unsigned u = (k & 0x80000000u) ? (k ^ 0x80000000u) : ~k;
  return __uint_as_float(u);
}

__device__ __forceinline__ void edge_e(const float* __restrict__ pl,
                                       const float* __restrict__ pr,
                                       const float* __restrict__ satt,
                                       float* eh) {
#pragma unroll
  for (int h = 0; h < Hq; ++h) {
    float s = 0.f;
#pragma unroll
    for (int d = 0; d < Dq; ++d) {
      float v = pl[h * Dq + d] + pr[h * Dq + d];
      v = v > 0.f ? v : 0.2f * v;           // leaky_relu(0.2)
      s += satt[h * Dq + d] * v;
    }
    eh[h] = s;
  }
}

// --------- Tensor Data Mover: 1-D tile (nelem f32) global -> LDS -----------
// D# per CDNA5 ISA: group0 = {count=1, lds_addr,
// global_addr[56:0], type=2}; group1 = {data_size=4B, tensor_dim0=tile_dim0=
// nelem, tensor_dim1=1}; groups 2/3 zero (<=2D tensor).
__device__ __forceinline__ void tdm_load_1d(const void* gsrc, unsigned lds_byte_off,
                                            unsigned nelem) {
  unsigned long long ga = (unsigned long long)gsrc;
  v4u g0;
  g0.x = 1u;                                                  // count=1
  g0.y = lds_byte_off;                                        // lds_addr
  g0.z = (unsigned)(ga & 0xFFFFFFFFu);                        // global_addr lo
  g0.w = (unsigned)((ga >> 32) & 0x01FFFFFFu) | (2u << 30);   // addr hi | type=2
  v8i g1;
  g1[0] = (int)(2u << 16);                                    // data_size = 4B
  g1[1] = (int)((nelem & 0xFFFFu) << 16);                     // tensor_dim0[15:0]
  g1[2] = (int)(((nelem >> 16) & 0xFFFFu) | (1u << 16));      // dim0[31:16]|dim1=1
  g1[3] = (int)((nelem & 0xFFFFu) << 16);                     // tile_dim0
  g1[4] = 0;                                                  // tile_dim1/2 = 0
  g1[5] = (int)nelem;                                         // dim0_stride lo
  g1[6] = 0;
  g1[7] = 0;
  v4i z4 = {0, 0, 0, 0};
#if __clang_major__ >= 23
  v8i z8 = {0, 0, 0, 0, 0, 0, 0, 0};
  __builtin_amdgcn_tensor_load_to_lds(g0, g1, z4, z4, z8, 0);
#else
  __builtin_amdgcn_tensor_load_to_lds(g0, g1, z4, z4, 0);
#endif
}

// LDS layout (dynamic): xl[N*HD] | xr[N*HD] | acc[N*HD] | emax[N*H] | esum[N*H] | satt[HD..]
#define SMEM_FLOATS (3 * Nq * HDq + 2 * Nq * Hq + 64)
#define SMEM_BYTES  (SMEM_FLOATS * sizeof(float))
#define ACC_LDS_BYTE_OFF (2u * Nq * HDq * 4u)   // dynamic LDS starts at 0 (no static LDS)

// ------------------------- GATv2 layer 0 (in = FIN=8) ----------------------
__global__ __launch_bounds__(NTHREADS)
void gat_layer0(const float* __restrict__ x,      // [B,N,FIN]
                const int*   __restrict__ ei,     // [2,E]
                const float* __restrict__ Wl,     // [FIN,HD]
                const float* __restrict__ Wr,     // [FIN,HD]
                const float* __restrict__ att,    // [H*D]
                const float* __restrict__ bias,   // [HD]
                float* __restrict__ h0,           // [B,N,HD]
                float* __restrict__ blockpart) {  // [B,40] sum/sumsq partials
  extern __shared__ float smem[];
  float*    xl   = smem;
  float*    xr   = xl + Nq * HDq;
  float*    acc  = xr + Nq * HDq;
  unsigned* emax = (unsigned*)(acc + Nq * HDq);
  float*    esum = (float*)(emax + Nq * Hq);
  float*    satt = esum + Nq * Hq;

  const int b = blockIdx.x;
  const int tid = threadIdx.x;
  const float* xb = x + (size_t)b * Nq * FINq;

  // warm the (shared across all 256 blocks) edge list into cache
  __builtin_prefetch(ei + tid * 8, 0, 1);
  __builtin_prefetch(ei + Eq + tid * 8, 0, 1);

  if (tid < HDq) satt[tid] = att[tid];
  // phase 1: projections + init
  for (int idx = tid; idx < Nq * HDq; idx += NTHREADS) {
    int n = idx / HDq, f = idx % HDq;
    float sl = 0.f, sr = 0.f;
    const float* xr_ = xb + n * FINq;
#pragma unroll
    for (int k = 0; k < FINq; ++k) {
      float v = xr_[k];
      sl += v * Wl[k * HDq + f];
      sr += v * Wr[k * HDq + f];
    }
    xl[idx] = sl; xr[idx] = sr; acc[idx] = 0.f;
  }
  for (int idx = tid; idx < Nq * Hq; idx += NTHREADS) { emax[idx] = 0u; esum[idx] = 0.f; }
  __syncthreads();

  // phase 2: segment max
  for (int e = tid; e < ESq; e += NTHREADS) {
    int s = (e < Eq) ? ei[e] : (e - Eq);
    int d = (e < Eq) ? ei[Eq + e] : (e - Eq);
    float eh[Hq];
    edge_e(&xl[s * HDq], &xr[d * HDq], satt, eh);
#pragma unroll
    for (int h = 0; h < Hq; ++h) atomicMax(&emax[d * Hq + h], fkey(eh[h]));
  }
  __syncthreads();

  // phase 3: segment sum of exp(e - max)
  for (int e = tid; e < ESq; e += NTHREADS) {
    int s = (e < Eq) ? ei[e] : (e - Eq);
    int d = (e < Eq) ? ei[Eq + e] : (e - Eq);
    float eh[Hq];
    edge_e(&xl[s * HDq], &xr[d * HDq], satt, eh);
#pragma unroll
    for (int h = 0; h < Hq; ++h)
      atomicAdd(&esum[d * Hq + h], __expf(eh[h] - finv(emax[d * Hq + h])));
  }
  __syncthreads();

  // phase 4: weighted scatter of alpha * xl[src]
  for (int e = tid; e < ESq; e += NTHREADS) {
    int s = (e < Eq) ? ei[e] : (e - Eq);
    int d = (e < Eq) ? ei[Eq + e] : (e - Eq);
    const float* pl = &xl[s * HDq];
    float eh[Hq];
    edge_e(pl, &xr[d * HDq], satt, eh);
#pragma unroll
    for (int h = 0; h < Hq; ++h) {
      float alpha = __expf(eh[h] - finv(emax[d * Hq + h])) / esum[d * Hq + h];
#pragma unroll
      for (int d2 = 0; d2 < Dq; ++d2)
        atomicAdd(&acc[d * HDq + h * Dq + d2], alpha * pl[h * Dq + d2]);
    }
  }
  __syncthreads();

  // phase 5: write h0 + per-block BN partials (sum, sumsq)
  float lsum[HDq], lsq[HDq];
#pragma unroll
  for (int f = 0; f < HDq; ++f) { lsum[f] = 0.f; lsq[f] = 0.f; }
  for (int n = tid; n < Nq; n += NTHREADS) {
    float* dst = h0 + ((size_t)b * Nq + n) * HDq;
#pragma unroll
    for (int f = 0; f < HDq; ++f) {
      float v = acc[n * HDq + f] + bias[f];
      dst[f] = v;
      lsum[f] += v; lsq[f] += v * v;
    }
  }
  float* part = xl;   // reuse (free after phase 4)
#pragma unroll
  for (int f = 0; f < HDq; ++f) { part[tid * 40 + f] = lsum[f]; part[tid * 40 + 20 + f] = lsq[f]; }
  __syncthreads();
  if (tid < 40) {
    float s = 0.f;
    for (int t = 0; t < NTHREADS; ++t) s += part[t * 40 + tid];
    blockpart[b * 40 + tid] = s;
  }
}

// ------------- BN finalize: scale = g*rsqrt(var+eps), shift = be - mu*scale --
__global__ void bn_finalize(const float* __restrict__ part,  // [B,40]
                            const float* __restrict__ g, const float* __restrict__ be,
                            float* __restrict__ ss) {        // [40] scale|shift
  __shared__ float tot[40];
  int t = threadIdx.x;
  if (t < 40) {
    float s = 0.f;
    for (int b = 0; b < Bq; ++b) s += part[b * 40 + t];
    tot[t] = s;
  }
  __syncthreads();
  if (t < HDq) {
    const float inv = 1.0f / (float)(Bq * Nq);
    float mu  = tot[t] * inv;
    float var = tot[20 + t] * inv - mu * mu;
    float sc  = rsqrtf(var + 1e-5f) * g[t];
    ss[t] = sc;
    ss[HDq + t] = be[t] - mu * sc;
  }
}

// ------------------------- GATv2 layer 1 (in = HD=20) ----------------------
__global__ __launch_bounds__(NTHREADS)
void gat_layer1(const float* __restrict__ h0,     // [B,N,HD] raw layer-0 out
                const int*   __restrict__ ei,
                const float* __restrict__ Wl,     // [HD,HD]
                const float* __restrict__ Wr,
                const float* __restrict__ att,
                const float* __restrict__ bias,
                const float* __restrict__ ss0,    // BN0 scale|shift
                float* __restrict__ blockpart,    // [B,40]
                float* __restrict__ nodesum) {    // [B,HD] sum over nodes of h1
  extern __shared__ float smem[];
  float*    xl   = smem;
  float*    xr   = xl + Nq * HDq;
  float*    acc  = xr + Nq * HDq;                 // staged input, later accumulator
  unsigned* emax = (unsigned*)(acc + Nq * HDq);
  float*    esum = (float*)(emax + Nq * Hq);
  float*    satt = esum + Nq * Hq;

  const int b = blockIdx.x;
  const int tid = threadIdx.x;
  const float* hb = h0 + (size_t)b * Nq * HDq;

  __builtin_prefetch(ei + tid * 8, 0, 1);
  __builtin_prefetch(ei + Eq + tid * 8, 0, 1);

  if (tid < HDq) satt[tid] = att[tid];

  // phase 1a: TDM-stage the raw 80KB [N,HD] batch tile of h0 into LDS 'acc'.
  // One tensor_load_to_lds issued by wave 0; TENSORcnt-tracked.
  if (tid < 32) {
    tdm_load_1d(hb, ACC_LDS_BYTE_OFF, Nq * HDq);
    __builtin_amdgcn_s_wait_tensorcnt(0);
  }
  __syncthreads();
  // phase 1a': BN0 affine applied in-place in LDS
  for (int idx = tid; idx < Nq * HDq; idx += NTHREADS) {
    int f = idx % HDq;
    acc[idx] = acc[idx] * ss0[f] + ss0[HDq + f];
  }
  __syncthreads();
  // phase 1b: projections from staged input
  for (int idx = tid; idx < Nq * HDq; idx += NTHREADS) {
    int n = idx / HDq, f = idx % HDq;
    float sl = 0.f, sr = 0.f;
    const float* in = &acc[n * HDq];
#pragma unroll
    for (int k = 0; k < HDq; ++k) {
      float v = in[k];
      sl += v * Wl[k * HDq + f];
      sr += v * Wr[k * HDq + f];
    }
    xl[idx] = sl; xr[idx] = sr;
  }
  __syncthreads();
  // phase 1c: reset accumulator + head state
  for (int idx = tid; idx < Nq * HDq; idx += NTHREADS) acc[idx] = 0.f;
  for (int idx = tid; idx < Nq * Hq; idx += NTHREADS) { emax[idx] = 0u; esum[idx] = 0.f; }
  __syncthreads();

  for (int e = tid; e < ESq; e += NTHREADS) {
    int s = (e < Eq) ? ei[e] : (e - Eq);
    int d = (e < Eq) ? ei[Eq + e] : (e - Eq);
    float eh[Hq];
    edge_e(&xl[s * HDq], &xr[d * HDq], satt, eh);
#pragma unroll
    for (int h = 0; h < Hq; ++h) atomicMax(&emax[d * Hq + h], fkey(eh[h]));
  }
  __syncthreads();

  for (int e = tid; e < ESq; e += NTHREADS) {
    int s = (e < Eq) ? ei[e] : (e - Eq);
    int d = (e < Eq) ? ei[Eq + e] : (e - Eq);
    float eh[Hq];
    edge_e(&xl[s * HDq], &xr[d * HDq], satt, eh);
#pragma unroll
    for (int h = 0; h < Hq; ++h)
      atomicAdd(&esum[d * Hq + h], __expf(eh[h] - finv(emax[d * Hq + h])));
  }
  __syncthreads();

  for (int e = tid; e < ESq; e += NTHREADS) {
    int s = (e < Eq) ? ei[e] : (e - Eq);
    int d = (e < Eq) ? ei[Eq + e] : (e - Eq);
    const float* pl = &xl[s * HDq];
    float eh[Hq];
    edge_e(pl, &xr[d * HDq], satt, eh);
#pragma unroll
    for (int h = 0; h < Hq; ++h) {
      float alpha = __expf(eh[h] - finv(emax[d * Hq + h])) / esum[d * Hq + h];
#pragma unroll
      for (int d2 = 0; d2 < Dq; ++d2)
        atomicAdd(&acc[d * HDq + h * Dq + d2], alpha * pl[h * Dq + d2]);
    }
  }
  __syncthreads();

  // phase 5: h1 is consumed here; only node sums + BN partials leave the WGP
  float lsum[HDq], lsq[HDq];
#pragma unroll
  for (int f = 0; f < HDq; ++f) { lsum[f] = 0.f; lsq[f] = 0.f; }
  for (int n = tid; n < Nq; n += NTHREADS) {
#pragma unroll
    for (int f = 0; f < HDq; ++f) {
      float v = acc[n * HDq + f] + bias[f];
      lsum[f] += v; lsq[f] += v * v;
    }
  }
  float* part = xl;
#pragma unroll
  for (int f = 0; f < HDq; ++f) { part[tid * 40 + f] = lsum[f]; part[tid * 40 + 20 + f] = lsq[f]; }
  __syncthreads();
  if (tid < 40) {
    float s = 0.f;
    for (int t = 0; t < NTHREADS; ++t) s += part[t * 40 + tid];
    blockpart[b * 40 + tid] = s;
    if (tid < HDq) nodesum[b * HDq + tid] = s;
  }
}

// ------- agg = bn1(mean_nodes(h1)) @ Wagg + bagg, then concat with obs ------
__global__ void agg_concat(const float* __restrict__ nodesum,  // [B,HD]
                           const float* __restrict__ ss1,      // BN1 scale|shift
                           const float* __restrict__ Wagg,     // [HD,64]
                           const float* __restrict__ bagg,     // [64]
                           const float* __restrict__ obs,      // [B,1024]
                           float* __restrict__ feat) {         // [B,1088]
  __shared__ float m[HDq];
  const int b = blockIdx.x, t = threadIdx.x;
  if (t < HDq) m[t] = nodesum[b * HDq + t] * (1.0f / Nq) * ss1[t] + ss1[HDq + t];
  __syncthreads();
  if (t < 64) {
    float s = bagg[t];
#pragma unroll
    for (int k = 0; k < HDq; ++k) s += m[k] * Wagg[k * 64 + t];
    feat[(size_t)b * 1088 + t] = s;
  }
  for (int j = t; j < 1024; j += blockDim.x)
    feat[(size_t)b * 1088 + 64 + j] = obs[(size_t)b * 1024 + j];
}

// ----------------- WMMA fp32 GEMM: C = act(A[M,K] @ W[K,N] + bias) ----------
// One wave per 16x16 tile, V_WMMA_F32_16X16X4_F32 along K.
// A layout per ISA: lanes 0-15 hold {K+0,K+1} of row M=lane; lanes 16-31 hold
// {K+2,K+3} of row M=lane-16. B mirrors with columns; C/D: vgpr r -> M=r(+8).
template <int ACT>
__global__ __launch_bounds__(128)
void gemm_wmma(const float* __restrict__ A, const float* __restrict__ W,
               const float* __restrict__ bias, float* __restrict__ C,
               int M, int N, int K) {
  const int wave = (blockIdx.x * blockDim.x + threadIdx.x) >> 5;
  const int lane = threadIdx.x & 31;
  const int tilesN = N >> 4;
  const int tiles  = (M >> 4) * tilesN;
  if (wave >= tiles) return;                  // wave-uniform
  const int tm = (wave / tilesN) << 4;
  const int tn = (wave % tilesN) << 4;
  const int half = lane >> 4;                 // 0: K+0..1, 1: K+2..3
  const int l    = lane & 15;

  v8f c = {};
  const float* arow = A + (size_t)(tm + l) * K + 2 * half;
  const float* wcol = W + (size_t)(2 * half) * N + tn + l;
  for (int k = 0; k < K; k += 4) {
    v2f a, bv;
    a.x  = arow[k];        a.y  = arow[k + 1];
    bv.x = wcol[(size_t)k * N]; bv.y = wcol[(size_t)(k + 1) * N];
    c = __builtin_amdgcn_wmma_f32_16x16x4_f32(false, a, false, bv, (short)0, c,
                                              false, false);
  }
  const int col = tn + l;
  const float bb = bias[col];
#pragma unroll
  for (int r = 0; r < 8; ++r) {
    float v = c[r] + bb;
    if (ACT) v = tanhf(v);
    C[(size_t)(tm + r + 8 * half) * N + col] = v;
  }
}

// ---------------------------------------------------------------------------
extern "C" void kernel_launch(void* const* d_in, const int* in_sizes, int n_in,
                              void* d_out, int out_size, void* d_ws, size_t ws_size,
                              hipStream_t stream) {
  const float* x    = (const float*)d_in[0];
  const float* obs  = (const float*)d_in[1];
  const int*   ei   = (const int*)  d_in[2];
  const float* Wl0  = (const float*)d_in[3];
  const float* Wr0  = (const float*)d_in[4];
  const float* att0 = (const float*)d_in[5];
  const float* b0   = (const float*)d_in[6];
  const float* Wl1  = (const float*)d_in[7];
  const float* Wr1  = (const float*)d_in[8];
  const float* att1 = (const float*)d_in[9];
  const float* b1   = (const float*)d_in[10];
  const float* g0   = (const float*)d_in[11];
  const float* be0  = (const float*)d_in[12];
  const float* g1   = (const float*)d_in[13];
  const float* be1  = (const float*)d_in[14];
  const float* Wagg = (const float*)d_in[15];
  const float* bagg = (const float*)d_in[16];
  const float* W1   = (const float*)d_in[17];
  const float* bh1  = (const float*)d_in[18];
  const float* W2   = (const float*)d_in[19];
  const float* bh2  = (const float*)d_in[20];
  const float* Wout = (const float*)d_in[21];
  const float* bout = (const float*)d_in[22];
  float* out = (float*)d_out;

  // workspace carve-up (floats)
  float* ws      = (float*)d_ws;
  float* h0      = ws;                         // B*N*HD
  float* part0   = h0 + (size_t)Bq * Nq * HDq; // B*40
  float* ss0     = part0 + Bq * 40;            // 40
  float* part1   = ss0 + 40;                   // B*40
  float* ss1     = part1 + Bq * 40;            // 40
  float* nodesum = ss1 + 40;                   // B*HD
  float* feat    = nodesum + Bq * HDq;         // B*1088
  float* mh1     = feat + (size_t)Bq * 1088;   // B*256
  float* mh2     = mh1 + (size_t)Bq * 256;     // B*128

  // 272KB dynamic LDS per block (CDNA5: up to 320KB/WGP)
  hipFuncSetAttribute((const void*)gat_layer0,
                      hipFuncAttributeMaxDynamicSharedMemorySize, (int)SMEM_BYTES);
  hipFuncSetAttribute((const void*)gat_layer1,
                      hipFuncAttributeMaxDynamicSharedMemorySize, (int)SMEM_BYTES);

  gat_layer0<<<Bq, NTHREADS, SMEM_BYTES, stream>>>(x, ei, Wl0, Wr0, att0, b0, h0, part0);
  bn_finalize<<<1, 64, 0, stream>>>(part0, g0, be0, ss0);
  gat_layer1<<<Bq, NTHREADS, SMEM_BYTES, stream>>>(h0, ei, Wl1, Wr1, att1, b1, ss0,
                                                   part1, nodesum);
  bn_finalize<<<1, 64, 0, stream>>>(part1, g1, be1, ss1);
  agg_concat<<<Bq, 128, 0, stream>>>(nodesum, ss1, Wagg, bagg, obs, feat);

  // MLP: [256,1088]@[1088,256] tanh -> [256,256]@[256,128] tanh -> [256,128]@[128,64]
  {
    int tiles = (256 / 16) * (256 / 16);   // 256 waves
    gemm_wmma<1><<<tiles * 32 / 128, 128, 0, stream>>>(feat, W1, bh1, mh1, 256, 256, 1088);
  }
  {
    int tiles = (256 / 16) * (128 / 16);   // 128 waves
    gemm_wmma<1><<<tiles * 32 / 128, 128, 0, stream>>>(mh1, W2, bh2, mh2, 256, 128, 256);
  }
  {
    int tiles = (256 / 16) * (64 / 16);    // 64 waves
    gemm_wmma<0><<<tiles * 32 / 128, 128, 0, stream>>>(mh2, Wout, bout, out, 256, 64, 128);
  }
}